// HydraGPU_27805618274844
// MI455X (gfx1250) — compile-verified
//
#include <hip/hip_runtime.h>
#include <math.h>

typedef __attribute__((ext_vector_type(2))) float v2f;
typedef __attribute__((ext_vector_type(8))) float v8f;

#define NEX     256      // examples
#define LIN     2048     // input length
#define NCH     256      // K*H output channels
#define KLEN    9        // conv taps
#define HGRP    32       // groups (H)
#define KK      8        // kernels per group (K)
#define NBRANCH 16       // 8 dilations * 2 (raw/diff)
#define NFEAT   8192     // NBRANCH * 2 * H * K
#define HALO    1024     // covers (tap-4)*d for d<=128 + 1-tile prefetch overrun

struct Bfrag { v2f b0, b1, b2; };

// One workgroup = one (example, branch). 16 waves; wave w owns channels
// [16w, 16w+16) = groups {2w, 2w+1}. Conv done as chained
// V_WMMA_F32_16X16X4_F32: A = 16ch x 4taps (W), B = 4taps x 16t (X patches).
// C-layout: lane l (<16) holds group 2w, time t0+l, kernels k=0..7 in acc[0..7];
// lanes 16-31 hold group 2w+1. So argmax/argmin over K=8 is lane-private.
__global__ __launch_bounds__(512)
void hydra_wmma_kernel(const float* __restrict__ X,
                       const float* __restrict__ W,
                       float* __restrict__ out) {
    __shared__ float xs[LIN + 2 * HALO];   // 16 KB, zero halo both sides

    const int n      = blockIdx.x;
    const int branch = blockIdx.y;         // di*2 + diff_index
    const int di     = branch >> 1;
    const int isdiff = branch & 1;
    const int d      = 1 << di;            // dilation 1..128

    const int tid   = threadIdx.x;
    const int wave  = tid >> 5;            // 0..15: channel tile
    const int lane  = tid & 31;
    const int tl    = lane & 15;           // time-in-tile / M-row index
    const int khalf = (lane >> 4) * 2;     // taps {0,1} (lanes 0-15) or {2,3}
    const float hi0 = (khalf == 0) ? 1.0f : 0.0f;  // mask for tap-8 chunk

    const int Tlen = isdiff ? (LIN - 1) : LIN;

    // ---- stage (diffed) input row into LDS with zero halo ----
    const float* xrow = X + (size_t)n * LIN;
    for (int i = tid; i < LIN + 2 * HALO; i += 512) {
        const int t = i - HALO;
        float v = 0.0f;
        if (t >= 0 && t < Tlen)
            v = isdiff ? (xrow[t + 1] - xrow[t]) : xrow[t];
        xs[i] = v;
    }
    __syncthreads();

    // ---- A fragments: 16 channels x 12 taps (taps 9..11 zero) ----
    // A 16x4 f32 layout: lanes0-15: M=lane, K=0(v0),K=1(v1); lanes16-31: K=2,K=3.
    const int ch = wave * 16 + tl;
    const float* wbr = W + ((size_t)branch * NCH + ch) * KLEN;
    v2f a0, a1, a2;
    a0.x = wbr[0 + khalf];  a0.y = wbr[1 + khalf];     // taps 0..3
    a1.x = wbr[4 + khalf];  a1.y = wbr[5 + khalf];     // taps 4..7
    a2.x = wbr[8] * hi0;    a2.y = 0.0f;               // tap 8, pad 9..11

    float cmax[KK], cmin[KK];
#pragma unroll
    for (int k = 0; k < KK; ++k) { cmax[k] = 0.0f; cmin[k] = 0.0f; }

    const float* xc = xs + HALO;           // centered pointer

    // Branch-free B-fragment load: all addresses stay inside the halo, so
    // loads are unconditional ds_load_b32; the tap-8 lane mask is a cndmask.
    auto loadB = [&](int t) {
        Bfrag f;
        f.b0.x = xc[t + (khalf - 4) * d];
        f.b0.y = xc[t + (khalf - 3) * d];
        f.b1.x = xc[t + (khalf + 0) * d];
        f.b1.y = xc[t + (khalf + 1) * d];
        f.b2.x = xc[t + 4 * d] * hi0;      // tap 8 only on lanes 0-15
        f.b2.y = 0.0f;
        return f;
    };

    // ---- time loop: 128 tiles of 16 output positions, software-pipelined ----
    Bfrag cur = loadB(tl);
    for (int t0 = 0; t0 < LIN; t0 += 16) {
        v8f acc = {};
        acc = __builtin_amdgcn_wmma_f32_16x16x4_f32(false, a0, false, cur.b0,
                                                    (short)0, acc, false, false);
        acc = __builtin_amdgcn_wmma_f32_16x16x4_f32(false, a1, false, cur.b1,
                                                    (short)0, acc, false, false);
        acc = __builtin_amdgcn_wmma_f32_16x16x4_f32(false, a2, false, cur.b2,
                                                    (short)0, acc, false, false);

        // prefetch next tile's B; LDS latency hidden under the reduction below
        // (last-iteration overrun lands in the zero halo: max idx < 4096)
        Bfrag nxt = loadB(t0 + 16 + tl);

        // lane-private reduction over the 8 kernels of this lane's group,
        // arithmetic time mask (only (diff, t==2047) is masked out)
        const float mask = (t0 + tl < Tlen) ? 1.0f : 0.0f;
        float bmax = acc[0], bmin = acc[0];
        int   imax = 0,      imin = 0;
#pragma unroll
        for (int k = 1; k < KK; ++k) {
            const float v = acc[k];
            if (v > bmax) { bmax = v; imax = k; }   // first-max (strict >)
            if (v < bmin) { bmin = v; imin = k; }   // first-min (strict <)
        }
        bmax *= mask;                               // masked-out lane adds +/-0
#pragma unroll
        for (int k = 0; k < KK; ++k) {
            cmax[k] += (imax == k) ? bmax : 0.0f;   // scatter-add max value
            cmin[k] += (imin == k) ? mask : 0.0f;   // count of mins
        }
        cur = nxt;
    }

    // ---- sum across the 16 lanes of each half (each half = one group) ----
#pragma unroll
    for (int k = 0; k < KK; ++k) {
#pragma unroll
        for (int m = 1; m <= 8; m <<= 1) {
            cmax[k] += __shfl_xor(cmax[k], m, 32);
            cmin[k] += __shfl_xor(cmin[k], m, 32);
        }
    }

    // lane 0 -> group 2w, lane 16 -> group 2w+1
    if ((lane & 15) == 0) {
        const int h = wave * 2 + (lane >> 4);
        float* obase = out + (size_t)n * NFEAT;
        const size_t fmax = ((size_t)(branch * 2 + 0) * HGRP + h) * KK;
        const size_t fmin = ((size_t)(branch * 2 + 1) * HGRP + h) * KK;
#pragma unroll
        for (int k = 0; k < KK; ++k) {
            obase[fmax + k] = sqrtf(fmaxf(cmax[k], 0.0f));  // safe_sqrt_clamp
            obase[fmin + k] = sqrtf(cmin[k]);               // counts >= 0
        }
    }
}

extern "C" void kernel_launch(void* const* d_in, const int* in_sizes, int n_in,
                              void* d_out, int out_size, void* d_ws, size_t ws_size,
                              hipStream_t stream) {
    const float* X = (const float*)d_in[0];   // [256, 1, 2048] f32
    const float* W = (const float*)d_in[1];   // [8, 2, 256, 1, 9] f32
    float* out = (float*)d_out;               // [256, 8192] f32

    dim3 grid(NEX, NBRANCH);                  // 4096 workgroups
    hydra_wmma_kernel<<<grid, 512, 0, stream>>>(X, W, out);
}